// MVDNet_nslpn_61632780698037
// MI455X (gfx1250) — compile-verified
//
#include <hip/hip_runtime.h>
#include <math.h>

#define NB   4
#define NH   480
#define NW   640
#define CHG  8
#define OC   24
#define OCP  25          // padded LDS row (conflict-free, gcd(25,64)=1)
#define KTOT 72          // 8 in-channels * 3*3
#define PROP_T 18
#define HWSZ (NH*NW)
#define BHW  (NB*HWSZ)

typedef __attribute__((ext_vector_type(2))) float v2f;
typedef __attribute__((ext_vector_type(8))) float v8f;

__device__ __forceinline__ float bilin_zero(const float* __restrict__ img, float y, float x) {
    // zero padding outside [0,H)x[0,W); clamp to avoid int overflow on wild offsets
    y = fminf(fmaxf(y, -1.0e6f), 1.0e6f);
    x = fminf(fmaxf(x, -1.0e6f), 1.0e6f);
    float y0f = floorf(y), x0f = floorf(x);
    float wy = y - y0f, wx = x - x0f;
    int y0 = (int)y0f, x0 = (int)x0f;
    int y1 = y0 + 1,   x1 = x0 + 1;
    bool vy0 = (y0 >= 0) && (y0 < NH);
    bool vy1 = (y1 >= 0) && (y1 < NH);
    bool vx0 = (x0 >= 0) && (x0 < NW);
    bool vx1 = (x1 >= 0) && (x1 < NW);
    float v00 = (vy0 && vx0) ? img[y0*NW + x0] : 0.0f;
    float v01 = (vy0 && vx1) ? img[y0*NW + x1] : 0.0f;
    float v10 = (vy1 && vx0) ? img[y1*NW + x0] : 0.0f;
    float v11 = (vy1 && vx1) ? img[y1*NW + x1] : 0.0f;
    float omwy = 1.0f - wy, omwx = 1.0f - wx;
    return v00*omwy*omwx + v01*omwy*wx + v10*wy*omwx + v11*wy*wx;
}

// --------------------------------------------------------------------------
// Setup: WMMA conv (8->24ch, 3x3) + tanh/TGASS + confidence modulation +
// abs-sum norm + softmax + tap coordinates. One wave handles 16 pixels.
// Grid covers B*H*W exactly => EXEC is all-ones at every WMMA.
// --------------------------------------------------------------------------
__global__ __launch_bounds__(256) void nslpn_setup(
    const float* __restrict__ guidance,    // (B,8,H,W)
    const float* __restrict__ confidence,  // (B,1,H,W)
    const float* __restrict__ w_oa,        // (24,8,3,3) -> flat (24,72)
    const float* __restrict__ b_oa,        // (24,)
    const float* __restrict__ aff_scale,   // (1,)
    float* __restrict__ ys8,               // (B,8,H,W) abs y coords (non-center taps)
    float* __restrict__ xs8,               // (B,8,H,W)
    float* __restrict__ aff9)              // (B,9,H,W) softmaxed affinities
{
    __shared__ __align__(16) float lds_w[32 * KTOT];   // weights padded to 32 rows
    __shared__ float lds_oa[128 * OCP];                // 128 px * 24ch (padded)

    const int tid = threadIdx.x;

    // Stage weights (zero-pad rows 24..31 for the second M-tile)
    for (int i = tid; i < 32 * KTOT; i += 256) {
        int m = i / KTOT, k = i - m * KTOT;
        lds_w[i] = (m < OC) ? w_oa[m * KTOT + k] : 0.0f;
    }
    __syncthreads();

    const int lane   = tid & 31;
    const int waveId = tid >> 5;
    const int n      = lane & 15;            // pixel column within tile (N)
    const int klo    = (lane >> 4) << 1;     // K sub-pair: 0 or 2
    const int pixRow = waveId * 16 + n;      // 0..127 within block
    const int p      = blockIdx.x * 128 + pixRow;

    const int b   = p / HWSZ;
    const int rem = p - b * HWSZ;
    const int py  = rem / NW;
    const int px  = rem - py * NW;

    const float* gb = guidance + (size_t)b * CHG * HWSZ;

    v8f acc0 = {0.f,0.f,0.f,0.f,0.f,0.f,0.f,0.f};   // channels 0..15
    v8f acc1 = {0.f,0.f,0.f,0.f,0.f,0.f,0.f,0.f};   // channels 16..31 (24..31 padded)

    #pragma unroll
    for (int kb = 0; kb < 18; ++kb) {
        const int k0 = kb * 4 + klo;
        // B-matrix: im2col values for this pixel, K = k0, k0+1
        v2f bb;
        {
            int k  = k0;
            int ci = k / 9, r9 = k - ci * 9;
            int gy = py + r9 / 3 - 1, gx = px + (r9 % 3) - 1;
            bb.x = (gy >= 0 && gy < NH && gx >= 0 && gx < NW)
                       ? gb[(ci * NH + gy) * NW + gx] : 0.0f;
            k  = k0 + 1;
            ci = k / 9; r9 = k - ci * 9;
            gy = py + r9 / 3 - 1; gx = px + (r9 % 3) - 1;
            bb.y = (gy >= 0 && gy < NH && gx >= 0 && gx < NW)
                       ? gb[(ci * NH + gy) * NW + gx] : 0.0f;
        }
        // A-matrix: weights, M = lane%16 (+16 for tile1), same K pair
        v2f a0 = *(const v2f*)&lds_w[ n        * KTOT + k0];
        v2f a1 = *(const v2f*)&lds_w[(n + 16)  * KTOT + k0];

        acc0 = __builtin_amdgcn_wmma_f32_16x16x4_f32(false, a0, false, bb,
                                                     (short)0, acc0, false, false);
        acc1 = __builtin_amdgcn_wmma_f32_16x16x4_f32(false, a1, false, bb,
                                                     (short)0, acc1, false, false);
    }

    // Scatter D tiles to LDS rows: lane holds N=lane%16, M = r + 8*(lane>>4)
    {
        float* myRow = &lds_oa[pixRow * OCP];
        const int chOfs = (lane >> 4) * 8;
        #pragma unroll
        for (int r = 0; r < 8; ++r) {
            int ch0 = r + chOfs;                 // 0..15
            myRow[ch0] = acc0[r] + b_oa[ch0];
            int ch1 = 16 + r + chOfs;            // 16..31
            if (ch1 < OC) myRow[ch1] = acc1[r] + b_oa[ch1];
        }
    }
    __syncthreads();

    // ---- scalar tail: one thread per pixel (first 128 threads) ----
    if (tid < 128) {
        const int pp   = blockIdx.x * 128 + tid;
        const int b2   = pp / HWSZ;
        const int rem2 = pp - b2 * HWSZ;
        const int y2   = rem2 / NW;
        const int x2   = rem2 - y2 * NW;

        const float* row = &lds_oa[tid * OCP];
        float oy[8], ox[8], a8[8];
        #pragma unroll
        for (int j = 0; j < 8; ++j) { oy[j] = row[j]; ox[j] = row[8 + j]; }

        const float inv_scale = 1.0f / (aff_scale[0] + 1e-8f);
        const float* cb = confidence + (size_t)b2 * HWSZ;

        float s = 0.0f;
        #pragma unroll
        for (int j = 0; j < 8; ++j) {
            float a = tanhf(row[16 + j]) * inv_scale;
            a *= bilin_zero(cb, (float)y2 + oy[j], (float)x2 + ox[j]);
            a8[j] = a;
            s += fabsf(a);
        }
        s += 1e-4f;
        s = fmaxf(s, 1.0f);
        const float inv_s = 1.0f / s;
        float tot = 0.0f;
        #pragma unroll
        for (int j = 0; j < 8; ++j) { a8[j] *= inv_s; tot += a8[j]; }

        float a9[9];
        a9[0] = a8[0]; a9[1] = a8[1]; a9[2] = a8[2]; a9[3] = a8[3];
        a9[4] = 1.0f - tot;
        a9[5] = a8[4]; a9[6] = a8[5]; a9[7] = a8[6]; a9[8] = a8[7];

        float mx = a9[0];
        #pragma unroll
        for (int t = 1; t < 9; ++t) mx = fmaxf(mx, a9[t]);
        float es = 0.0f;
        #pragma unroll
        for (int t = 0; t < 9; ++t) { a9[t] = expf(a9[t] - mx); es += a9[t]; }
        const float inv_es = 1.0f / es;

        const float khv[9] = {-1,-1,-1, 0,0,0, 1,1,1};
        const float kwv[9] = {-1, 0, 1,-1,0,1,-1,0,1};
        const int yx = y2 * NW + x2;

        #pragma unroll
        for (int t = 0; t < 9; ++t)
            aff9[(size_t)(b2 * 9 + t) * HWSZ + yx] = a9[t] * inv_es;

        #pragma unroll
        for (int tt = 0; tt < 8; ++tt) {
            int t = (tt < 4) ? tt : tt + 1;       // skip center tap
            ys8[(size_t)(b2 * 8 + tt) * HWSZ + yx] = (float)y2 + khv[t] + oy[tt];
            xs8[(size_t)(b2 * 8 + tt) * HWSZ + yx] = (float)x2 + kwv[t] + ox[tt];
        }
    }
}

// --------------------------------------------------------------------------
// One propagation iteration: center tap is an exact read, 8 bilinear gathers.
// --------------------------------------------------------------------------
__global__ __launch_bounds__(256) void nslpn_prop(
    const float* __restrict__ src, float* __restrict__ dst,
    const float* __restrict__ ys8, const float* __restrict__ xs8,
    const float* __restrict__ aff9)
{
    const int p = blockIdx.x * 256 + threadIdx.x;
    if (p >= BHW) return;
    const int b   = p / HWSZ;
    const int rem = p - b * HWSZ;

    const float* img = src + (size_t)b * HWSZ;
    float acc = aff9[(size_t)(b * 9 + 4) * HWSZ + rem] * img[rem];

    #pragma unroll
    for (int tt = 0; tt < 8; ++tt) {
        const int t9 = (tt < 4) ? tt : tt + 1;
        float y = ys8[(size_t)(b * 8 + tt) * HWSZ + rem];
        float x = xs8[(size_t)(b * 8 + tt) * HWSZ + rem];
        float a = aff9[(size_t)(b * 9 + t9) * HWSZ + rem];
        acc += a * bilin_zero(img, y, x);
    }
    dst[p] = acc;
}

extern "C" void kernel_launch(void* const* d_in, const int* in_sizes, int n_in,
                              void* d_out, int out_size, void* d_ws, size_t ws_size,
                              hipStream_t stream) {
    const float* feat_init  = (const float*)d_in[0];
    const float* guidance   = (const float*)d_in[1];
    const float* confidence = (const float*)d_in[2];
    const float* w_oa       = (const float*)d_in[3];
    const float* b_oa       = (const float*)d_in[4];
    const float* aff_scale  = (const float*)d_in[5];
    float* out = (float*)d_out;

    float* ws  = (float*)d_ws;
    float* ys8  = ws;                         // 8*BHW
    float* xs8  = ys8 + (size_t)8 * BHW;      // 8*BHW
    float* aff9 = xs8 + (size_t)8 * BHW;      // 9*BHW
    float* fA   = aff9 + (size_t)9 * BHW;     // BHW
    float* fB   = fA + (size_t)BHW;           // BHW

    // 1,228,800 pixels / 128 per block = 9600 blocks (exact fit, EXEC all-ones)
    nslpn_setup<<<BHW / 128, 256, 0, stream>>>(guidance, confidence, w_oa, b_oa,
                                               aff_scale, ys8, xs8, aff9);

    const float* src = feat_init;
    for (int t = 0; t < PROP_T; ++t) {
        float* dst = (t == PROP_T - 1) ? out : ((t & 1) ? fB : fA);
        nslpn_prop<<<BHW / 256, 256, 0, stream>>>(src, dst, ys8, xs8, aff9);
        src = dst;
    }
}